// RobustQuoteModule_38362647888258
// MI455X (gfx1250) — compile-verified
//
#include <hip/hip_runtime.h>
#include <hip/hip_bf16.h>
#include <math.h>

// ---------------------------------------------------------------------------
// dims
#define NCANCH 10
#define NH     6
#define CDIM   384
#define HD     64
#define BB     16
#define TT     197
#define NN     196
#define NP     208     // padded N (13*16)
#define PITCH  212     // LDS row pitch (floats): 16B-aligned rows, bank-clean
#define KC     49      // 196/4 contraction chunks (padded chunks are never read)
#define SCALE  0.125f

typedef __attribute__((ext_vector_type(2))) float v2f;
typedef __attribute__((ext_vector_type(8))) float v8f;

__device__ __forceinline__ v8f wmma4(v2f a, v2f b, v8f c) {
  // D(16x16 f32) = A(16x4 f32) * B(4x16 f32) + C
  return __builtin_amdgcn_wmma_f32_16x16x4_f32(false, a, false, b, (short)0, c,
                                               false, false);
}
__device__ __forceinline__ v8f v8zero() {
  v8f z;
#pragma unroll
  for (int i = 0; i < 8; ++i) z[i] = 0.0f;
  return z;
}
__device__ __forceinline__ float gelu_exact(float v) {
  return 0.5f * v * (1.0f + erff(v * 0.70710678118654752f));
}

// ---------------------------------------------------------------------------
// LayerNorm: one block (128 thr) per row of 384.  t==0 -> gA/bA else gB/bB.
__global__ void ln_kernel(const float* __restrict__ x,
                          const float* __restrict__ gA, const float* __restrict__ bA,
                          const float* __restrict__ gB, const float* __restrict__ bB,
                          float* __restrict__ out, int tokens) {
  __shared__ float s1[128], s2[128];
  int row = blockIdx.x;
  int tid = threadIdx.x;
  const float* xr = x + (long long)row * CDIM;
  float a0 = xr[tid], a1 = xr[tid + 128], a2 = xr[tid + 256];
  s1[tid] = a0 + a1 + a2;
  s2[tid] = a0 * a0 + a1 * a1 + a2 * a2;
  __syncthreads();
  for (int o = 64; o > 0; o >>= 1) {
    if (tid < o) { s1[tid] += s1[tid + o]; s2[tid] += s2[tid + o]; }
    __syncthreads();
  }
  float mu = s1[0] * (1.0f / CDIM);
  float var = s2[0] * (1.0f / CDIM) - mu * mu;
  float rs = rsqrtf(var + 1e-5f);
  bool cls = (row % tokens) == 0;
  const float* g = cls ? gA : gB;
  const float* be = cls ? bA : bB;
  float* orow = out + (long long)row * CDIM;
  orow[tid]       = (a0 - mu) * rs * g[tid]       + be[tid];
  orow[tid + 128] = (a1 - mu) * rs * g[tid + 128] + be[tid + 128];
  orow[tid + 256] = (a2 - mu) * rs * g[tid + 256] + be[tid + 256];
}

// ---------------------------------------------------------------------------
// Generic fp32 WMMA GEMM:  out = act(Xb @ W + bias) [+ addsrc]
// One wave computes a 16x64 tile (4 accumulators; A frag reused 4x).
// grid = (cols/64, ceil(rows/16), batch)
// mode 0: out[b*oBatch + m*oRow + n]
// mode 1: head-major out[b*oBatch + ((n>>6)*rows + m)*64 + (n&63)]
__global__ void gemm_wmma(const float* __restrict__ X, long long xBatch, int xRow,
                          int rows, int K,
                          const float* __restrict__ W, int cols,
                          const float* __restrict__ bias, int act,
                          const float* __restrict__ addsrc, long long aBatch, int aRow,
                          float* __restrict__ out, long long oBatch, int oRow, int mode) {
  int b = blockIdx.z;
  int rt = blockIdx.y, ct = blockIdx.x;
  int lane = threadIdx.x & 31;
  int m16 = lane & 15, hi = lane >> 4;
  int khalf = hi << 1;
  // clamp A row: out-of-range A rows only affect discarded output rows
  int mA = rt * 16 + m16;
  if (mA > rows - 1) mA = rows - 1;
  const float* xp = X + (long long)b * xBatch + (long long)mA * xRow + khalf;
  const float* wp = W + (long long)khalf * cols + ct * 64 + m16;
  const long long wstep = (long long)cols * 4;
  v8f facc[4];
#pragma unroll
  for (int j = 0; j < 4; ++j) facc[j] = v8zero();
  for (int kk = 0; kk < K; kk += 4) {
    v2f a = *(const v2f*)xp;
    v2f bf[4];
#pragma unroll
    for (int j = 0; j < 4; ++j) {
      bf[j].x = wp[j * 16];
      bf[j].y = wp[cols + j * 16];
    }
#pragma unroll
    for (int j = 0; j < 4; ++j) facc[j] = wmma4(a, bf[j], facc[j]);
    xp += 4;
    wp += wstep;
  }
#pragma unroll
  for (int j = 0; j < 4; ++j) {
    int n = ct * 64 + j * 16 + m16;
    float bv = bias ? bias[n] : 0.0f;
#pragma unroll
    for (int r = 0; r < 8; ++r) {
      int m = rt * 16 + r + 8 * hi;
      if (m >= rows) continue;
      float v = facc[j][r] + bv;
      if (act) v = gelu_exact(v);
      if (addsrc) v += addsrc[(long long)b * aBatch + (long long)m * aRow + n];
      long long off;
      if (mode == 0) off = (long long)b * oBatch + (long long)m * oRow + n;
      else off = (long long)b * oBatch + (((long long)(n >> 6) * rows + m) << 6) + (n & 63);
      out[off] = v;
    }
  }
}

// ---------------------------------------------------------------------------
// cls attention (tiny): logits, softmax over anchors, weighted v.
// qv: (10,768) [q | v]; k: (16,384); attn out: [b][n*6+h]; cin: [b][h*64+d]
__global__ void cls_attn_kernel(const float* __restrict__ qv,
                                const float* __restrict__ k,
                                float* __restrict__ attn, float* __restrict__ cin) {
  int t = threadIdx.x;
  if (t >= BB * NH) return;
  int b = t / NH, h = t % NH;
  float e[NCANCH];
  float mx = -1e30f;
  for (int n = 0; n < NCANCH; ++n) {
    float s = 0.0f;
    for (int d = 0; d < HD; ++d)
      s += qv[n * 768 + h * HD + d] * k[b * CDIM + h * HD + d];
    e[n] = s * SCALE;
    mx = fmaxf(mx, e[n]);
  }
  float Z = 0.0f;
  for (int n = 0; n < NCANCH; ++n) { e[n] = expf(e[n] - mx); Z += e[n]; }
  float iZ = 1.0f / Z;
  for (int n = 0; n < NCANCH; ++n) {
    e[n] *= iZ;
    attn[b * (NCANCH * NH) + n * NH + h] = e[n];
  }
  for (int d = 0; d < HD; ++d) {
    float s = 0.0f;
    for (int n = 0; n < NCANCH; ++n)
      s += e[n] * qv[n * 768 + CDIM + h * HD + d];
    cin[b * CDIM + h * HD + d] = s;
  }
}

// ---------------------------------------------------------------------------
// Fused pairwise-attention core.  One block per (b,h); 8 waves.
// Per anchor c: R^T = relu(S * K2_c @ Q2_b^T) built transposed in LDS
// (sRT[k][m], so L@R B-frags are single b64 LDS loads); each wave builds its
// 16x208 row-tile L = relu(-S * Qh_b @ Kh_c^T) in a private LDS slab;
// acc += w2[b,h,c] * (L @ R).  Epilogue: rect = tmp @ vr.
// All contraction loops run only KC=49 chunks (k<196); padded regions are
// written but never read, so no guards/selects are needed in hot loops.
__global__ __launch_bounds__(256) void pair_attn_kernel(
    const float* __restrict__ qh, const float* __restrict__ q2,
    const float* __restrict__ kh, const float* __restrict__ k2,
    const float* __restrict__ vr, const float* __restrict__ attn,
    float* __restrict__ rectpre) {
  extern __shared__ float smem[];
  float* sRT = smem;                      // NP * PITCH  (R transposed: [k][m])
  float* sLall = smem + NP * PITCH;       // 8 * 16 * PITCH
  int b = blockIdx.x / NH, h = blockIdx.x % NH;
  int wave = threadIdx.x >> 5, lane = threadIdx.x & 31;
  int m16 = lane & 15, hi = lane >> 4;
  int khalf = hi << 1;
  const long long hoff = (long long)(b * NH + h) * NN * HD;
  const float* qh_b = qh + hoff;
  const float* q2_b = q2 + hoff;
  const float* vr_b = vr + hoff;
  float* sL = sLall + wave * 16 * PITCH;

  for (int pass = 0; pass < 2; ++pass) {
    int nt = pass * 8 + wave;
    bool active = nt < 13;
    v8f acc[13];
#pragma unroll
    for (int kt = 0; kt < 13; ++kt) acc[kt] = v8zero();

    for (int c = 0; c < NCANCH; ++c) {
      const float* kh_c = kh + (long long)(c * NH + h) * NN * HD;
      const float* k2_c = k2 + (long long)(c * NH + h) * NN * HD;
      __syncthreads();
      // ---- build R^T (169 tiles split across 8 waves); rows clamped ----
      for (int t = wave; t < 169; t += 8) {
        int mt = t / 13, kt = t % 13;
        int mrow = mt * 16 + m16;
        if (mrow > NN - 1) mrow = NN - 1;
        int ncol = kt * 16 + m16;
        int ncl = ncol > NN - 1 ? NN - 1 : ncol;
        const float* ap = k2_c + mrow * HD + khalf;
        const float* bp = q2_b + ncl * HD + khalf;
        v8f f = v8zero();
#pragma unroll
        for (int d = 0; d < HD; d += 4) {
          v2f a = *(const v2f*)(ap + d);
          v2f bf = *(const v2f*)(bp + d);
          f = wmma4(a, bf, f);
        }
        float* dst = sRT + ncol * PITCH + mt * 16 + 8 * hi;
        float4 o0, o1;
        {
          float v0 = SCALE * f[0], v1 = SCALE * f[1], v2 = SCALE * f[2], v3 = SCALE * f[3];
          float v4 = SCALE * f[4], v5 = SCALE * f[5], v6 = SCALE * f[6], v7 = SCALE * f[7];
          o0.x = v0 > 0.f ? v0 : 0.f; o0.y = v1 > 0.f ? v1 : 0.f;
          o0.z = v2 > 0.f ? v2 : 0.f; o0.w = v3 > 0.f ? v3 : 0.f;
          o1.x = v4 > 0.f ? v4 : 0.f; o1.y = v5 > 0.f ? v5 : 0.f;
          o1.z = v6 > 0.f ? v6 : 0.f; o1.w = v7 > 0.f ? v7 : 0.f;
        }
        *(float4*)dst = o0;
        *(float4*)(dst + 4) = o1;
      }
      __syncthreads();
      if (active) {
        // ---- build L row tile (16 x 208) into private LDS slab ----
        int nrow = nt * 16 + m16;
        if (nrow > NN - 1) nrow = NN - 1;
        const float* ap0 = qh_b + nrow * HD + khalf;
        for (int mt = 0; mt < 13; ++mt) {
          int mcol = mt * 16 + m16;
          int mc = mcol > NN - 1 ? NN - 1 : mcol;
          const float* bp = kh_c + mc * HD + khalf;
          v8f f = v8zero();
#pragma unroll
          for (int d = 0; d < HD; d += 4) {
            v2f a = *(const v2f*)(ap0 + d);
            v2f bf = *(const v2f*)(bp + d);
            f = wmma4(a, bf, f);
          }
#pragma unroll
          for (int r = 0; r < 8; ++r) {
            float v = -SCALE * f[r];
            sL[(r + 8 * hi) * PITCH + mcol] = v > 0.0f ? v : 0.0f;
          }
        }
        __threadfence_block();
        // w2[b,h,c] via flat reinterpretation of attn_cls (B,NC,H)->(B,H,NC)
        float w2v = attn[b * (NCANCH * NH) + h * NCANCH + c];
        // ---- acc += w2 * (L @ R) : pure b64 LDS loads + wmma ----
        const float* arow = sL + m16 * PITCH + khalf;
#pragma unroll
        for (int kt = 0; kt < 13; ++kt) {
          const float* brow = sRT + (kt * 16 + m16) * PITCH + khalf;
          v8f f = v8zero();
#pragma unroll 7
          for (int q = 0; q < KC; ++q) {
            v2f a = *(const v2f*)(arow + q * 4);
            v2f bf = *(const v2f*)(brow + q * 4);
            f = wmma4(a, bf, f);
          }
#pragma unroll
          for (int r = 0; r < 8; ++r) acc[kt][r] += w2v * f[r];
        }
      }
    }
    // ---- epilogue: rect_pre[b][h][nt rows][d] = tmp @ vr ----
    if (active) {
#pragma unroll
      for (int kt = 0; kt < 13; ++kt) {
        int col = kt * 16 + m16;
#pragma unroll
        for (int r = 0; r < 8; ++r)
          sL[(r + 8 * hi) * PITCH + col] = acc[kt][r];
      }
      __threadfence_block();
      const float* arow = sL + m16 * PITCH + khalf;
      for (int dt = 0; dt < 4; ++dt) {
        const float* bp = vr_b + khalf * HD + dt * 16 + m16;
        v8f f = v8zero();
#pragma unroll 7
        for (int q = 0; q < KC; ++q) {
          v2f a = *(const v2f*)(arow + q * 4);
          v2f bf;
          bf.x = bp[q * 4 * HD];
          bf.y = bp[q * 4 * HD + HD];
          f = wmma4(a, bf, f);
        }
#pragma unroll
        for (int r = 0; r < 8; ++r) {
          int n = nt * 16 + r + 8 * hi;
          if (n < NN)
            rectpre[hoff + (long long)n * HD + dt * 16 + m16] = f[r];
        }
      }
    }
  }
}

// ---------------------------------------------------------------------------
extern "C" void kernel_launch(void* const* d_in, const int* in_sizes, int n_in,
                              void* d_out, int out_size, void* d_ws, size_t ws_size,
                              hipStream_t stream) {
  (void)in_sizes; (void)n_in; (void)out_size; (void)ws_size;
  const float* x       = (const float*)d_in[0];
  const float* Wqv     = (const float*)d_in[1];
  const float* Wk      = (const float*)d_in[2];
  const float* Wq_proj = (const float*)d_in[3];
  const float* bq_proj = (const float*)d_in[4];
  const float* Wqk     = (const float*)d_in[5];
  const float* Wqk2    = (const float*)d_in[6];
  const float* Wv      = (const float*)d_in[7];
  const float* Wr_proj = (const float*)d_in[8];
  const float* br_proj = (const float*)d_in[9];
  const float* g1 = (const float*)d_in[10], *b1 = (const float*)d_in[11];
  const float* g2 = (const float*)d_in[12], *b2 = (const float*)d_in[13];
  const float* g3 = (const float*)d_in[14], *b3 = (const float*)d_in[15];
  const float* W1  = (const float*)d_in[16], *bm1 = (const float*)d_in[17];
  const float* W2  = (const float*)d_in[18], *bm2 = (const float*)d_in[19];
  float* out = (float*)d_out;

  // workspace layout (floats)
  float* ws    = (float*)d_ws;
  float* lnbuf = ws;                              // 26*197*384
  float* qvbuf = lnbuf + 26LL * TT * CDIM;        // 10*768
  float* kbuf  = qvbuf + NCANCH * 768;            // 16*384
  float* attnb = kbuf + BB * CDIM;                // 16*60
  float* cinb  = attnb + BB * NCANCH * NH;        // 16*384
  float* qhb   = cinb + BB * CDIM;                // 16*6*196*64
  float* q2b   = qhb + (long long)BB * NH * NN * HD;
  float* vrb   = q2b + (long long)BB * NH * NN * HD;
  float* khb   = vrb + (long long)BB * NH * NN * HD;     // 10*6*196*64
  float* k2b   = khb + (long long)NCANCH * NH * NN * HD;
  float* rpre  = k2b + (long long)NCANCH * NH * NN * HD; // 16*196*384
  float* xxb   = rpre + (long long)BB * NN * CDIM;       // 16*197*384
  float* ln2b  = xxb + (long long)BB * TT * CDIM;        // 16*197*384
  float* h1b   = ln2b + (long long)BB * TT * CDIM;       // 16*197*1536

  const long long rowT = (long long)TT * CDIM;     // 197*384
  const long long hB   = (long long)NH * NN * HD;  // 75264

  // 1) LN over all 26*197 rows (g1/b1 on cls token, g2/b2 on image tokens)
  ln_kernel<<<26 * TT, 128, 0, stream>>>(x, g1, b1, g2, b2, lnbuf, TT);

  // 2) qv = n1a @ Wqv (10x768);  k = n1x @ Wk (16x384)
  gemm_wmma<<<dim3(12, 1, NCANCH), 32, 0, stream>>>(
      lnbuf + 16LL * rowT, rowT, CDIM, 1, CDIM, Wqv, 768,
      nullptr, 0, nullptr, 0, 0, qvbuf, 768LL, 768, 0);
  gemm_wmma<<<dim3(6, 1, BB), 32, 0, stream>>>(
      lnbuf, rowT, CDIM, 1, CDIM, Wk, CDIM,
      nullptr, 0, nullptr, 0, 0, kbuf, (long long)CDIM, CDIM, 0);

  // 3) cls attention softmax + weighted v
  cls_attn_kernel<<<1, 128, 0, stream>>>(qvbuf, kbuf, attnb, cinb);

  // 4) cls_out = cin @ Wq_proj + bq_proj  -> xx rows t=0 (oRow = 197*384)
  gemm_wmma<<<dim3(6, 1, 1), 32, 0, stream>>>(
      cinb, 0LL, CDIM, BB, CDIM, Wq_proj, CDIM,
      bq_proj, 0, nullptr, 0, 0, xxb, 0LL, (int)rowT, 0);

  // 5) head projections (mode 1 stores [b][h][n][d])
  gemm_wmma<<<dim3(6, 13, BB), 32, 0, stream>>>(
      lnbuf + CDIM, rowT, CDIM, NN, CDIM, Wqk, CDIM,
      nullptr, 0, nullptr, 0, 0, qhb, hB, 0, 1);
  gemm_wmma<<<dim3(6, 13, BB), 32, 0, stream>>>(
      lnbuf + CDIM, rowT, CDIM, NN, CDIM, Wqk2, CDIM,
      nullptr, 0, nullptr, 0, 0, q2b, hB, 0, 1);
  gemm_wmma<<<dim3(6, 13, BB), 32, 0, stream>>>(
      lnbuf + CDIM, rowT, CDIM, NN, CDIM, Wv, CDIM,
      nullptr, 0, nullptr, 0, 0, vrb, hB, 0, 1);
  gemm_wmma<<<dim3(6, 13, NCANCH), 32, 0, stream>>>(
      lnbuf + 16LL * rowT + CDIM, rowT, CDIM, NN, CDIM, Wqk, CDIM,
      nullptr, 0, nullptr, 0, 0, khb, hB, 0, 1);
  gemm_wmma<<<dim3(6, 13, NCANCH), 32, 0, stream>>>(
      lnbuf + 16LL * rowT + CDIM, rowT, CDIM, NN, CDIM, Wqk2, CDIM,
      nullptr, 0, nullptr, 0, 0, k2b, hB, 0, 1);

  // 6) fused pairwise attention core (285 KB LDS of the 320 KB WGP pool)
  const int SMEM_BYTES = (NP * PITCH + 8 * 16 * PITCH) * (int)sizeof(float); // 284928
  hipFuncSetAttribute((const void*)pair_attn_kernel,
                      hipFuncAttributeMaxDynamicSharedMemorySize, SMEM_BYTES);
  pair_attn_kernel<<<BB * NH, 256, SMEM_BYTES, stream>>>(
      qhb, q2b, khb, k2b, vrb, attnb, rpre);

  // 7) rect = rect_pre @ Wr_proj + br_proj + x_im  -> xx rows t>=1
  gemm_wmma<<<dim3(6, 13, BB), 32, 0, stream>>>(
      rpre, hB, CDIM, NN, CDIM, Wr_proj, CDIM,
      br_proj, 0, x + CDIM, rowT, CDIM, xxb + CDIM, rowT, CDIM, 0);

  // 8) LN(xx) with g3/b3 on every row
  ln_kernel<<<BB * TT, 128, 0, stream>>>(xxb, g3, b3, g3, b3, ln2b, TT);

  // 9) MLP: h1 = gelu(ln2 @ W1 + bm1)   (3152 x 1536)
  gemm_wmma<<<dim3(24, 197, 1), 32, 0, stream>>>(
      ln2b, 0LL, CDIM, BB * TT, CDIM, W1, 1536,
      bm1, 1, nullptr, 0, 0, h1b, 0LL, 1536, 0);

  // 10) out[:16] = xx + h1 @ W2 + bm2
  gemm_wmma<<<dim3(6, 197, 1), 32, 0, stream>>>(
      h1b, 0LL, 1536, BB * TT, 1536, W2, CDIM,
      bm2, 0, xxb, 0LL, CDIM, out, 0LL, CDIM, 0);

  // 11) anchors pass through unchanged
  hipMemcpyAsync(out + 16LL * rowT, x + 16LL * rowT,
                 (size_t)NCANCH * TT * CDIM * sizeof(float),
                 hipMemcpyDeviceToDevice, stream);
}